// fism_79525614452999
// MI455X (gfx1250) — compile-verified
//
#include <hip/hip_runtime.h>

#define BATCH 4096
#define HIST 200
#define NNEG 20
#define DIM 64
#define BATA 1e-4f
#define LAMDA 1e-4f

typedef __attribute__((ext_vector_type(2))) float v2f;
typedef __attribute__((ext_vector_type(8))) float v8f;

__device__ __forceinline__ float wave_sum(float v) {
#pragma unroll
  for (int m = 16; m > 0; m >>= 1) v += __shfl_xor(v, m, 32);
  return v;
}

// One wave (32 lanes) per user; 16 users (waves) per block.
__global__ __launch_bounds__(512) void fism_main(
    const int* __restrict__ pos_items,
    const int* __restrict__ neg_items,
    const float* __restrict__ user_item_num,
    const int* __restrict__ interacted,
    const float* __restrict__ pu,
    const float* __restrict__ qi,
    const float* __restrict__ bi,
    float* __restrict__ partial) {
  __shared__ float u_lds[16][DIM];    // per-wave user embedding
  __shared__ float scores[16][24];    // per-wave 21 scores (pos, neg0..19)

  const int wave = threadIdx.x >> 5;   // 0..15
  const int lane = threadIdx.x & 31;   // wave32
  const int user = blockIdx.x * 16 + wave;

  // ---------------- embedding bag: u = sum of 200 pu rows ----------------
  // Each iteration issues 4 independent, fully coalesced 256B row loads.
  const int* hist = interacted + user * HIST;
  float ax0 = 0.f, ay0 = 0.f, ax1 = 0.f, ay1 = 0.f;
  float ax2 = 0.f, ay2 = 0.f, ax3 = 0.f, ay3 = 0.f;
  for (int h = 0; h < HIST; h += 4) {
    const int i0 = hist[h];
    const int i1 = hist[h + 1];
    const int i2 = hist[h + 2];
    const int i3 = hist[h + 3];
    if (h + 19 < HIST) {
      __builtin_prefetch(pu + (size_t)hist[h + 16] * DIM + lane * 2, 0, 3);
      __builtin_prefetch(pu + (size_t)hist[h + 17] * DIM + lane * 2, 0, 3);
      __builtin_prefetch(pu + (size_t)hist[h + 18] * DIM + lane * 2, 0, 3);
      __builtin_prefetch(pu + (size_t)hist[h + 19] * DIM + lane * 2, 0, 3);
    }
    const float2 x0 = ((const float2*)(pu + (size_t)i0 * DIM))[lane];
    const float2 x1 = ((const float2*)(pu + (size_t)i1 * DIM))[lane];
    const float2 x2 = ((const float2*)(pu + (size_t)i2 * DIM))[lane];
    const float2 x3 = ((const float2*)(pu + (size_t)i3 * DIM))[lane];
    ax0 += x0.x; ay0 += x0.y;
    ax1 += x1.x; ay1 += x1.y;
    ax2 += x2.x; ay2 += x2.y;
    ax3 += x3.x; ay3 += x3.y;
  }
  float2 u2;
  u2.x = (ax0 + ax1) + (ax2 + ax3);
  u2.y = (ay0 + ay1) + (ay2 + ay3);
  ((float2*)u_lds[wave])[lane] = u2;
  const float usq = wave_sum(u2.x * u2.x + u2.y * u2.y); // sum(user_embeds^2)
  __syncthreads();

  // ---------------- scores via V_WMMA_F32_16X16X4_F32 ----------------
  // A (16x4 f32): lane l holds row (l&15), cols 4c + 2*(l>>4) .. +1
  //   pass1 rows: 0 = pos item, 1..15 = neg 0..14
  //   pass2 rows: 0..4 = neg 15..19, 5..15 = zero pad (cndmask-selected)
  // B (4x16 f32): user vector replicated across N -> lane l holds
  //   u[4c + 2*(l>>4)], u[4c + 2*(l>>4) + 1]   (LDS broadcast)
  // D: with replicated N, lane0 c[j]=score[j] (j<8), lane16 c[j]=score[8+j].
  const int r = lane & 15;
  const int half = lane >> 4;
  const bool live2 = (r < 5);

  const int item1 = (r == 0) ? pos_items[user]
                             : neg_items[user * NNEG + (r - 1)];
  // Pad lanes read qi row 0 (always valid, hot in cache) and select 0 below:
  // straight-line cndmask instead of exec-mask branches around each load.
  const int item2 = live2 ? neg_items[user * NNEG + 15 + r] : 0;
  const float* e1ptr = qi + (size_t)item1 * DIM + 2 * half;
  const float* e2ptr = qi + (size_t)item2 * DIM + 2 * half;

  v8f c1 = {0.f, 0.f, 0.f, 0.f, 0.f, 0.f, 0.f, 0.f};
  v8f c2 = {0.f, 0.f, 0.f, 0.f, 0.f, 0.f, 0.f, 0.f};
  float esq = 0.f;  // sum(pos_embeds^2) + sum(neg_embeds^2), over real rows
#pragma unroll
  for (int c = 0; c < 16; ++c) {
    const float2 a1 = *(const float2*)(e1ptr + 4 * c);
    const float2 a2raw = *(const float2*)(e2ptr + 4 * c);
    const float a2x = live2 ? a2raw.x : 0.f;
    const float a2y = live2 ? a2raw.y : 0.f;
    const float2 bu = *(const float2*)(&u_lds[wave][4 * c + 2 * half]);
    v2f av1; av1[0] = a1.x; av1[1] = a1.y;
    v2f av2; av2[0] = a2x;  av2[1] = a2y;
    v2f bv;  bv[0]  = bu.x; bv[1]  = bu.y;
    c1 = __builtin_amdgcn_wmma_f32_16x16x4_f32(false, av1, false, bv,
                                               (short)0, c1, false, false);
    c2 = __builtin_amdgcn_wmma_f32_16x16x4_f32(false, av2, false, bv,
                                               (short)0, c2, false, false);
    esq += a1.x * a1.x + a1.y * a1.y + a2x * a2x + a2y * a2y;
  }
  esq = wave_sum(esq);

  if (lane == 0) {
#pragma unroll
    for (int j = 0; j < 8; ++j) scores[wave][j] = c1[j];
#pragma unroll
    for (int j = 0; j < 5; ++j) scores[wave][16 + j] = c2[j];
  }
  if (lane == 16) {
#pragma unroll
    for (int j = 0; j < 8; ++j) scores[wave][8 + j] = c1[j];
  }
  __syncthreads();

  // ---------------- per-user loss ----------------
  const float t = rsqrtf(user_item_num[user]);  // n^(-0.5), ALPHA = 0.5
  const float b_pos = bi[pos_items[user]];
  const float pos_score = t * scores[wave][0] + b_pos;

  float dsum = 0.f, bnsq = 0.f;
  if (lane < NNEG) {
    const float bn = bi[neg_items[user * NNEG + lane]];
    const float neg_score = t * scores[wave][1 + lane] + bn;
    const float diff = 1.0f - pos_score + neg_score;  // 1 - (pos - neg)
    dsum = diff * diff;
    bnsq = bn * bn;
  }
  dsum = wave_sum(dsum);
  bnsq = wave_sum(bnsq);

  if (lane == 0) {
    partial[user] = dsum + BATA * usq + BATA * esq +
                    LAMDA * (b_pos * b_pos + bnsq);
  }
}

// Deterministic final reduction of 4096 per-user partials -> scalar.
__global__ __launch_bounds__(256) void fism_reduce(
    const float* __restrict__ partial, float* __restrict__ out, int n) {
  __shared__ float s[256];
  float a = 0.f;
  for (int i = threadIdx.x; i < n; i += 256) a += partial[i];
  s[threadIdx.x] = a;
  __syncthreads();
#pragma unroll
  for (int m = 128; m > 0; m >>= 1) {
    if ((int)threadIdx.x < m) s[threadIdx.x] += s[threadIdx.x + m];
    __syncthreads();
  }
  if (threadIdx.x == 0) out[0] = s[0];
}

extern "C" void kernel_launch(void* const* d_in, const int* in_sizes, int n_in,
                              void* d_out, int out_size, void* d_ws,
                              size_t ws_size, hipStream_t stream) {
  const int* pos_items = (const int*)d_in[0];
  const int* neg_items = (const int*)d_in[1];
  const float* user_item_num = (const float*)d_in[2];
  const int* interacted = (const int*)d_in[3];
  const float* pu = (const float*)d_in[4];
  const float* qi = (const float*)d_in[5];
  const float* bi = (const float*)d_in[6];

  float* partial = (float*)d_ws;  // BATCH floats of scratch

  fism_main<<<BATCH / 16, 512, 0, stream>>>(pos_items, neg_items,
                                            user_item_num, interacted, pu, qi,
                                            bi, partial);
  fism_reduce<<<1, 256, 0, stream>>>(partial, (float*)d_out, BATCH);
}